// SQLGNNEncoder_88923002896888
// MI455X (gfx1250) — compile-verified
//
#include <hip/hip_runtime.h>

typedef __attribute__((ext_vector_type(2))) float v2f;
typedef __attribute__((ext_vector_type(8))) float v8f;

#define FDIM 32
#define TB 256

__device__ __forceinline__ void atomAddF(float* p, float v) {
    unsafeAtomicAdd(p, v);   // hardware global_atomic_add_f32
}

// ---------------- init / utility ----------------

__global__ void k_init(float* __restrict__ acc, float* __restrict__ deg,
                       float* __restrict__ pool, float* __restrict__ cnt,
                       int N, int G) {
    int idx = blockIdx.x * blockDim.x + threadIdx.x;
    if (idx < N * FDIM) acc[idx] = 0.f;
    if (idx < N)        deg[idx] = 1.f;      // self-loop contributes 1 to degree
    if (idx < G * FDIM) pool[idx] = 0.f;
    if (idx < G)        cnt[idx] = 0.f;
}

__global__ void k_zero(float* __restrict__ p, int n) {
    int idx = blockIdx.x * blockDim.x + threadIdx.x;
    if (idx < n) p[idx] = 0.f;
}

__global__ void k_deg(const long long* __restrict__ col, float* __restrict__ deg, int E) {
    int e = blockIdx.x * blockDim.x + threadIdx.x;
    if (e < E) atomAddF(&deg[(int)col[e]], 1.f);
}

__global__ void k_rsqrt(float* __restrict__ deg, int N) {
    int i = blockIdx.x * blockDim.x + threadIdx.x;
    if (i < N) deg[i] = rsqrtf(deg[i]);      // deg >= 1 always (self-loops)
}

// ---------------- dense linears via V_WMMA_F32_16X16X4_F32 ----------------
// One wave -> one 16(M) x 16(N) fp32 output tile.
// A 16x4 layout: lane L: m=L&15, kbase=(L>>4)*2; a={A[m][kbase],A[m][kbase+1]}
// B 4x16 layout: lane L: n=L&15, kbase=(L>>4)*2; b={B[kbase][n],B[kbase+1][n]}
// C/D: c[v] = D[v + (L>=16?8:0)][L&15]

// x:[N,2] @ W1:[2,32] -> out:[N,32].  K padded 2->4 with zeros (lanes 16..31).
__global__ void k_gemm1_wmma(const float* __restrict__ x, const float* __restrict__ W1,
                             float* __restrict__ out, int N, int nwaves) {
    int wave = (blockIdx.x * blockDim.x + threadIdx.x) >> 5;
    if (wave >= nwaves) return;                 // wave-uniform exit
    int lane = threadIdx.x & 31;
    int tile = wave >> 1;
    int n0   = (wave & 1) * 16;
    int m    = lane & 15;
    int hi   = lane >> 4;                       // 0: K=0,1   1: K=2,3 (zero pad)
    int row  = tile * 16 + m;

    v2f b = {0.f, 0.f};
    if (hi == 0) { b.x = W1[n0 + m]; b.y = W1[FDIM + n0 + m]; }

    if (tile * 16 + 16 <= N) {                  // wave-uniform full-tile path
        v2f av = *(const v2f*)(x + row * 2);    // b64, coalesced
        v2f a;
        a.x = hi ? 0.f : av.x;
        a.y = hi ? 0.f : av.y;
        v8f c = {};
        c = __builtin_amdgcn_wmma_f32_16x16x4_f32(false, a, false, b, (short)0, c,
                                                  false, false);
        int rbase = tile * 16 + hi * 8;
#pragma unroll
        for (int v = 0; v < 8; ++v)
            out[(rbase + v) * FDIM + n0 + m] = c[v];
    } else {                                    // partial edge tile (rare)
        v2f a = {0.f, 0.f};
        if (hi == 0 && row < N) { a.x = x[row * 2 + 0]; a.y = x[row * 2 + 1]; }
        v8f c = {};
        c = __builtin_amdgcn_wmma_f32_16x16x4_f32(false, a, false, b, (short)0, c,
                                                  false, false);
        int rbase = tile * 16 + hi * 8;
#pragma unroll
        for (int v = 0; v < 8; ++v) {
            int r = rbase + v;
            if (r < N) out[r * FDIM + n0 + m] = c[v];
        }
    }
}

// h:[N,32] @ W2:[32,32] -> out:[N,32].  8 chained K=4 WMMAs.
__global__ void k_gemm2_wmma(const float* __restrict__ h, const float* __restrict__ W2,
                             float* __restrict__ out, int N, int nwaves) {
    int wave = (blockIdx.x * blockDim.x + threadIdx.x) >> 5;
    if (wave >= nwaves) return;
    int lane = threadIdx.x & 31;
    int tile = wave >> 1;
    int n0   = (wave & 1) * 16;
    int m    = lane & 15;
    int hi   = lane >> 4;
    int row  = tile * 16 + m;

    if (tile * 16 + 16 <= N) {                  // wave-uniform full-tile path
        v8f c = {};
#pragma unroll
        for (int k8 = 0; k8 < 8; ++k8) {
            int k = k8 * 4 + hi * 2;
            v2f a = *(const v2f*)(h + row * FDIM + k);   // b64, coalesced
            v2f b;
            b.x = W2[k * FDIM + n0 + m];
            b.y = W2[(k + 1) * FDIM + n0 + m];
            c = __builtin_amdgcn_wmma_f32_16x16x4_f32(false, a, false, b, (short)0, c,
                                                      false, false);
        }
        int rbase = tile * 16 + hi * 8;
#pragma unroll
        for (int v = 0; v < 8; ++v)
            out[(rbase + v) * FDIM + n0 + m] = c[v];
    } else {                                    // partial edge tile (rare)
        bool rok = row < N;
        v8f c = {};
#pragma unroll
        for (int k8 = 0; k8 < 8; ++k8) {
            int k = k8 * 4 + hi * 2;
            v2f a = {0.f, 0.f};
            if (rok) { a.x = h[row * FDIM + k]; a.y = h[row * FDIM + k + 1]; }
            v2f b;
            b.x = W2[k * FDIM + n0 + m];
            b.y = W2[(k + 1) * FDIM + n0 + m];
            c = __builtin_amdgcn_wmma_f32_16x16x4_f32(false, a, false, b, (short)0, c,
                                                      false, false);
        }
        int rbase = tile * 16 + hi * 8;
#pragma unroll
        for (int v = 0; v < 8; ++v) {
            int r = rbase + v;
            if (r < N) out[r * FDIM + n0 + m] = c[v];
        }
    }
}

// ---------------- sparse propagate: one wave per edge, lane per feature ----
// Edge indices are wave-uniform -> readfirstlane so index math / dinv loads
// scalarize; per-lane work is a coalesced 128B gather + 128B of f32 atomics.
__global__ void k_scatter(const long long* __restrict__ ei,   // [2E]: src rows then dst cols
                          const float* __restrict__ dinv,
                          const float* __restrict__ t,
                          float* __restrict__ acc, int E) {
    int idx = blockIdx.x * blockDim.x + threadIdx.x;
    int e = idx >> 5;                  // all 32 lanes of a wave share one edge
    if (e >= E) return;
    int j = idx & 31;
    int r = __builtin_amdgcn_readfirstlane((int)ei[e]);
    int c = __builtin_amdgcn_readfirstlane((int)ei[E + e]);
    float nrm = dinv[r] * dinv[c];     // uniform -> scalar loads
    atomAddF(&acc[c * FDIM + j], nrm * t[r * FDIM + j]);
}

// acc = relu(acc + dinv^2 * t + b)   (self-loop edge folded in analytically)
__global__ void k_final1(float4* __restrict__ acc, const float4* __restrict__ t,
                         const float* __restrict__ dinv, const float4* __restrict__ b1,
                         int N) {
    int idx = blockIdx.x * blockDim.x + threadIdx.x;     // over N*8 float4s
    if (idx >= N * (FDIM / 4)) return;
    int i  = idx >> 3;
    int j4 = idx & 7;
    float d = dinv[i];
    float d2 = d * d;
    float4 a = acc[idx], tv = t[idx], bb = b1[j4];
    float4 o;
    o.x = fmaxf(a.x + d2 * tv.x + bb.x, 0.f);
    o.y = fmaxf(a.y + d2 * tv.y + bb.y, 0.f);
    o.z = fmaxf(a.z + d2 * tv.z + bb.z, 0.f);
    o.w = fmaxf(a.w + d2 * tv.w + bb.w, 0.f);
    acc[idx] = o;
}

// layer-2 finalize + pooled sums (sorted batch -> L2-local atomics)
__global__ void k_final2_pool(const float4* __restrict__ acc, const float4* __restrict__ t,
                              const float* __restrict__ dinv, const float4* __restrict__ b2,
                              const long long* __restrict__ batch,
                              float* __restrict__ pool, float* __restrict__ cnt, int N) {
    int idx = blockIdx.x * blockDim.x + threadIdx.x;     // over N*8 float4s
    if (idx >= N * (FDIM / 4)) return;
    int i  = idx >> 3;
    int j4 = idx & 7;
    float d = dinv[i];
    float d2 = d * d;
    float4 a = acc[idx], tv = t[idx], bb = b2[j4];
    int g = (int)batch[i];
    float* pg = pool + g * FDIM + j4 * 4;
    atomAddF(pg + 0, a.x + d2 * tv.x + bb.x);
    atomAddF(pg + 1, a.y + d2 * tv.y + bb.y);
    atomAddF(pg + 2, a.z + d2 * tv.z + bb.z);
    atomAddF(pg + 3, a.w + d2 * tv.w + bb.w);
    if (j4 == 0) atomAddF(&cnt[g], 1.f);
}

__global__ void k_pool_div(const float* __restrict__ pool, const float* __restrict__ cnt,
                           float* __restrict__ out, int G) {
    int idx = blockIdx.x * blockDim.x + threadIdx.x;
    if (idx >= G * FDIM) return;
    out[idx] = pool[idx] / fmaxf(cnt[idx >> 5], 1.f);
}

// ---------------- launch ----------------

extern "C" void kernel_launch(void* const* d_in, const int* in_sizes, int n_in,
                              void* d_out, int out_size, void* d_ws, size_t ws_size,
                              hipStream_t stream) {
    const float*     x     = (const float*)d_in[0];
    const long long* ei    = (const long long*)d_in[1];   // [2, E] int64
    const long long* batch = (const long long*)d_in[2];   // [N] int64
    // d_in[3] = num_graphs scalar (derived from out_size instead)
    const float*     W1    = (const float*)d_in[4];       // [2, 32]
    const float*     b1    = (const float*)d_in[5];       // [32]
    const float*     W2    = (const float*)d_in[6];       // [32, 32]
    const float*     b2    = (const float*)d_in[7];       // [32]
    float*           out   = (float*)d_out;

    int N = in_sizes[0] / 2;
    int E = in_sizes[1] / 2;
    int G = out_size / FDIM;

    size_t Na = ((size_t)N + 3) & ~(size_t)3;      // keep 16B alignment below
    float* deg  = (float*)d_ws;                    // N      (deg -> dinv in place)
    float* B1   = deg + Na;                        // 32N    (t1, then t2)
    float* B2   = B1 + (size_t)N * FDIM;           // 32N    (h1acc/h1, then h2acc)
    float* pool = B2 + (size_t)N * FDIM;           // 32G
    float* cnt  = pool + (size_t)G * FDIM;         // G

    int totNF   = N * FDIM;
    int nfBlks  = (totNF + TB - 1) / TB;
    int n4      = N * (FDIM / 4);
    int n4Blks  = (n4 + TB - 1) / TB;
    int tiles   = (N + 15) / 16;
    int nwaves  = tiles * 2;                       // 2 column halves of 16
    int gBlks   = (nwaves * 32 + TB - 1) / TB;
    int sBlks   = (int)(((long long)E * 32 + TB - 1) / TB);

    k_init  <<<nfBlks, TB, 0, stream>>>(B2, deg, pool, cnt, N, G);
    k_deg   <<<(E + TB - 1) / TB, TB, 0, stream>>>(ei + (size_t)E, deg, E);
    k_rsqrt <<<(N + TB - 1) / TB, TB, 0, stream>>>(deg, N);

    // layer 1
    k_gemm1_wmma <<<gBlks, TB, 0, stream>>>(x, W1, B1, N, nwaves);
    k_scatter    <<<sBlks, TB, 0, stream>>>(ei, deg, B1, B2, E);
    k_final1     <<<n4Blks, TB, 0, stream>>>((float4*)B2, (const float4*)B1, deg,
                                             (const float4*)b1, N);

    // layer 2
    k_gemm2_wmma <<<gBlks, TB, 0, stream>>>(B2, W2, B1, N, nwaves);
    k_zero       <<<nfBlks, TB, 0, stream>>>(B2, totNF);
    k_scatter    <<<sBlks, TB, 0, stream>>>(ei, deg, B1, B2, E);

    // pool
    k_final2_pool <<<n4Blks, TB, 0, stream>>>((const float4*)B2, (const float4*)B1, deg,
                                              (const float4*)b2, batch, pool, cnt, N);
    k_pool_div    <<<(G * FDIM + TB - 1) / TB, TB, 0, stream>>>(pool, cnt, out, G);
}